// BiLinearInteractionLayer_19679540150798
// MI455X (gfx1250) — compile-verified
//
#include <hip/hip_runtime.h>

typedef __attribute__((ext_vector_type(2))) float v2f;
typedef __attribute__((ext_vector_type(4))) float v4f;
typedef __attribute__((ext_vector_type(8))) float v8f;

constexpr int FN     = 40;     // fields
constexpr int DIM    = 64;     // embed dim
constexpr int NPAIR  = 780;    // FN*(FN-1)/2
constexpr int MPAD   = 48;     // 40 padded to 3x16 WMMA tiles
constexpr int LSTR   = 68;     // LDS row stride (floats), conflict-free, 16B aligned

__global__ __launch_bounds__(256)
void bilinear_wmma_kernel(const float* __restrict__ inputs,
                          const float* __restrict__ w,
                          float* __restrict__ out)
{
    __shared__ float sA [MPAD * LSTR];   // inputs[b], zero-padded rows 40..47
    __shared__ float sW [DIM  * LSTR];   // shared weight
    __shared__ float sXW[MPAD * LSTR];   // inputs[b] @ W
    __shared__ short sPI[NPAIR];
    __shared__ short sPJ[NPAIR];

    const int b   = blockIdx.x;
    const int tid = threadIdx.x;

    // ---- Stage A: stage inputs[b] and W into LDS (128-bit loads) ----
    const v4f* gin = (const v4f*)(inputs + (size_t)b * FN * DIM);
    for (int q = tid; q < FN * (DIM / 4); q += 256) {
        int row = q >> 4, c4 = q & 15;
        *(v4f*)&sA[row * LSTR + c4 * 4] = gin[q];
    }
    for (int q = tid; q < (MPAD - FN) * (DIM / 4); q += 256) {   // zero pad rows
        int row = FN + (q >> 4), c4 = q & 15;
        *(v4f*)&sA[row * LSTR + c4 * 4] = (v4f){0.f, 0.f, 0.f, 0.f};
    }
    const v4f* gw = (const v4f*)w;
    for (int q = tid; q < DIM * (DIM / 4); q += 256) {
        int row = q >> 4, c4 = q & 15;
        *(v4f*)&sW[row * LSTR + c4 * 4] = gw[q];
    }
    // pair index tables (itertools.combinations order)
    for (int p = tid; p < NPAIR; p += 256) {
        int i = 0, rem = p;
        while (rem >= (FN - 1 - i)) { rem -= (FN - 1 - i); ++i; }
        sPI[p] = (short)i;
        sPJ[p] = (short)(i + 1 + rem);
    }
    __syncthreads();

    // ---- Stage B: XW[48x64] = A[48x64] * W[64x64] via v_wmma_f32_16x16x4_f32 ----
    // 12 tiles (3 M x 4 N) over 8 waves; wave-uniform loop keeps EXEC all-ones.
    const int wave = tid >> 5;
    const int lane = tid & 31;
    const int half = lane >> 4;   // selects K pair within the 16x4 A fragment
    const int mr   = lane & 15;

    for (int t = wave; t < 12; t += 8) {
        const int mt   = t >> 2;
        const int nt   = t & 3;
        const int arow = (mt * 16 + mr) * LSTR;
        const int bcol = nt * 16 + mr;
        v8f acc = {};
        #pragma unroll
        for (int k0 = 0; k0 < DIM; k0 += 4) {
            const int k = k0 + 2 * half;
            // A 16x4 fragment: lane M = mr, VGPR v holds K = 2*half + v (contiguous -> b64 LDS load)
            v2f av = *(const v2f*)&sA[arow + k];
            // B 4x16 fragment: lane N = mr, VGPR v holds K = 2*half + v (rows LSTR apart)
            v2f bv;
            bv.x = sW[(k    ) * LSTR + bcol];
            bv.y = sW[(k + 1) * LSTR + bcol];
            acc = __builtin_amdgcn_wmma_f32_16x16x4_f32(
                      /*neg_a=*/false, av, /*neg_b=*/false, bv,
                      /*c_mod=*/(short)0, acc,
                      /*reuse_a=*/false, /*reuse_b=*/false);
        }
        // C/D layout: VGPR r, lane-half selects M = mt*16 + r + 8*half, N = bcol
        #pragma unroll
        for (int r = 0; r < 8; ++r)
            sXW[(mt * 16 + r + 8 * half) * LSTR + bcol] = acc[r];
    }
    __syncthreads();

    // ---- Stage C: pair products, 128-bit non-temporal coalesced stores ----
    v4f* gout = (v4f*)out + (size_t)b * (NPAIR * (DIM / 4));
    for (int e = tid; e < NPAIR * (DIM / 4); e += 256) {
        int p  = e >> 4;
        int c4 = (e & 15) * 4;
        int i  = sPI[p], j = sPJ[p];
        v4f x = *(const v4f*)&sXW[i * LSTR + c4];
        v4f y = *(const v4f*)&sA [j * LSTR + c4];
        v4f r = x * y;
        __builtin_nontemporal_store(r, &gout[e]);   // output never re-read: keep L2 for W
    }
}

extern "C" void kernel_launch(void* const* d_in, const int* in_sizes, int n_in,
                              void* d_out, int out_size, void* d_ws, size_t ws_size,
                              hipStream_t stream)
{
    const float* inputs = (const float*)d_in[0];
    const float* w      = (const float*)d_in[1];
    float*       out    = (float*)d_out;
    const int B = in_sizes[0] / (FN * DIM);   // 2048
    bilinear_wmma_kernel<<<B, 256, 0, stream>>>(inputs, w, out);
}